// DetectionTargetLayer_25580825215451
// MI455X (gfx1250) — compile-verified
//
#include <hip/hip_runtime.h>
#include <stdint.h>
#include <math.h>

// ---------------- reference constants ----------------
#define TT    200          // T
#define PP    66           // int(T * 0.33)
#define TN    134          // T - P
#define NPROP 2000
#define NGT   64
#define HH    512
#define WW    512
#define MHH   28
#define MWW   28
#define EPSF  1e-8f

// =====================================================================
// Kernel 1: IoU row max + argmax.  GT boxes/ids staged in LDS (1.25 KB).
// 4 blocks x 256 threads (8 wave32 waves / block).
// =====================================================================
__global__ void k_iou(const float* __restrict__ props,
                      const int*   __restrict__ gids,
                      const float* __restrict__ gboxes,
                      float* __restrict__ iou_max,
                      int*   __restrict__ assign_out) {
  __shared__ float sb[NGT][4];
  __shared__ int   sid[NGT];
  const int b = blockIdx.x;
  for (int g = threadIdx.x; g < NGT; g += blockDim.x) {
    const float* gb = gboxes + ((size_t)b * NGT + g) * 4;
    sb[g][0] = gb[0]; sb[g][1] = gb[1]; sb[g][2] = gb[2]; sb[g][3] = gb[3];
    sid[g] = gids[b * NGT + g];
  }
  __syncthreads();

  for (int i = threadIdx.x; i < NPROP; i += blockDim.x) {
    const float* p = props + ((size_t)b * NPROP + i) * 4;
    if (i + (int)blockDim.x < NPROP)
      __builtin_prefetch(p + 4 * blockDim.x, 0, 0);   // gfx1250 global_prefetch_b8
    const float p0 = p[0], p1 = p[1], p2 = p[2], p3 = p[3];
    const float a1 = (p2 - p0) * (p3 - p1);
    float best = -1.0f; int bi = 0;
    for (int g = 0; g < NGT; ++g) {
      float v = -1.0f;
      if (sid[g] > 0) {
        const float y1 = fmaxf(p0, sb[g][0]);
        const float x1 = fmaxf(p1, sb[g][1]);
        const float y2 = fminf(p2, sb[g][2]);
        const float x2 = fminf(p3, sb[g][3]);
        const float inter = fmaxf(y2 - y1, 0.0f) * fmaxf(x2 - x1, 0.0f);
        const float a2 = (sb[g][2] - sb[g][0]) * (sb[g][3] - sb[g][1]);
        v = inter / fmaxf(a1 + a2 - inter, EPSF);
      }
      if (v > best) { best = v; bi = g; }   // first-max wins == jnp.argmax
    }
    iou_max[b * NPROP + i] = best;
    assign_out[b * NPROP + i] = bi;
  }
}

// =====================================================================
// Kernel 2: stable top-k selection via exact rank counting (matches
// lax.top_k tie-breaking: lower index first).  IoU row (8 KB) staged in
// LDS with CDNA5 async global->LDS (ASYNCcnt) instead of load+ds_store.
// =====================================================================
__global__ void k_select(const float* __restrict__ iou_max,
                         int*   __restrict__ posidx,
                         float* __restrict__ posval,
                         int*   __restrict__ negidx,
                         int*   __restrict__ negn) {
  __shared__ float s[NPROP];
  __shared__ int s_np, s_nn;
  const int b = blockIdx.x;
  if (threadIdx.x == 0) { s_np = 0; s_nn = 0; }
  const float* row = iou_max + (size_t)b * NPROP;

  {
    // Async DMA: 500 x b128 lane-transfers, LDS address = low 32 bits of
    // the flat shared pointer (allocation-relative LDS offset).
    const unsigned lbase = (unsigned)(uintptr_t)(void*)s;
    for (int i = threadIdx.x; i < NPROP / 4; i += blockDim.x) {
      const unsigned loff = lbase + (unsigned)i * 16u;
      const unsigned goff = (unsigned)i * 16u;    // GVS: saddr base + vgpr offset
      asm volatile("global_load_async_to_lds_b128 %0, %1, %2"
                   :: "v"(loff), "v"(goff), "s"(row) : "memory");
    }
#if defined(__has_builtin) && __has_builtin(__builtin_amdgcn_s_wait_asynccnt)
    __builtin_amdgcn_s_wait_asynccnt(0);
#else
    asm volatile("s_wait_asynccnt 0" ::: "memory");
#endif
  }
  __syncthreads();

  // Pass 1: population counts (pos_count / neg_mask.sum()).
  int lp = 0, ln = 0;
  for (int i = threadIdx.x; i < NPROP; i += blockDim.x) {
    const float m = s[i];
    if (m >= 0.5f) lp++;
    if (m <  0.5f) ln++;
  }
  atomicAdd(&s_np, lp);
  atomicAdd(&s_nn, ln);
  __syncthreads();
  const int npos = s_np, nneg = s_nn;

  // Pass 2: exact ranks (injective over a strict total order, so every
  // slot rank<k is written exactly once -> deterministic, stable top-k).
  for (int i = threadIdx.x; i < NPROP; i += blockDim.x) {
    const float m = s[i];
    const float ai = (m >= 0.5f) ? m  : -1.0f;        // positive score
    const float bi = (m <  0.5f) ? -m : -1.0e9f;      // negative score
    int rp = 0, rn = 0;
    for (int j = 0; j < NPROP; ++j) {
      const float mj = s[j];
      const float aj = (mj >= 0.5f) ? mj  : -1.0f;
      const float bj = (mj <  0.5f) ? -mj : -1.0e9f;
      rp += (aj > ai) || (aj == ai && j < i);
      rn += (bj > bi) || (bj == bi && j < i);
    }
    if (rp < PP) { posidx[b * PP + rp] = i; posval[b * PP + rp] = ai; }
    if (rn < TN) { negidx[b * TN + rn] = i; }
  }

  if (threadIdx.x == 0) {
    const int pc = npos < PP ? npos : PP;             // pos_valid.sum()
    int nt = (int)((float)pc / 0.33f) - pc;           // neg_target
    nt = nt < 0 ? 0 : (nt > TN ? TN : nt);
    negn[b] = nt < nneg ? nt : nneg;                  // neg_n
  }
}

// =====================================================================
// Kernel 3: rois / class_ids / deltas (and their zero tails).
// =====================================================================
__global__ void k_outputs(const float* __restrict__ props,
                          const int*   __restrict__ gids,
                          const float* __restrict__ gboxes,
                          const int*   __restrict__ posidx,
                          const float* __restrict__ posval,
                          const int*   __restrict__ negidx,
                          const int*   __restrict__ negn,
                          const int*   __restrict__ assign_in,
                          float* __restrict__ rois,
                          float* __restrict__ cls,
                          float* __restrict__ deltas) {
  const int b = blockIdx.x;
  const int t = threadIdx.x;
  if (t >= TT) return;

  float r0 = 0.f, r1 = 0.f, r2 = 0.f, r3 = 0.f;
  float d0 = 0.f, d1 = 0.f, d2 = 0.f, d3 = 0.f;
  float c  = 0.f;

  if (t < PP) {
    const int   i     = posidx[b * PP + t];
    const float valid = (posval[b * PP + t] >= 0.5f) ? 1.0f : 0.0f;
    const float* p = props + ((size_t)b * NPROP + i) * 4;
    const float p0 = p[0], p1 = p[1], p2 = p[2], p3 = p[3];
    r0 = p0 * valid; r1 = p1 * valid; r2 = p2 * valid; r3 = p3 * valid;

    const int a = assign_in[b * NPROP + i];
    c = valid * (float)gids[b * NGT + a];

    const float* gb = gboxes + ((size_t)b * NGT + a) * 4;
    const float h  = fmaxf(p2 - p0, EPSF);
    const float w  = fmaxf(p3 - p1, EPSF);
    const float cy = p0 + 0.5f * h;
    const float cx = p1 + 0.5f * w;
    const float gh = fmaxf(gb[2] - gb[0], EPSF);
    const float gw = fmaxf(gb[3] - gb[1], EPSF);
    const float gcy = gb[0] + 0.5f * gh;
    const float gcx = gb[1] + 0.5f * gw;
    d0 = (((gcy - cy) / h) / 0.1f) * valid;
    d1 = (((gcx - cx) / w) / 0.1f) * valid;
    d2 = (logf(gh / h) / 0.2f) * valid;
    d3 = (logf(gw / w) / 0.2f) * valid;
  } else {
    const int tn = t - PP;
    const int j  = negidx[b * TN + tn];
    const float nv = (tn < negn[b]) ? 1.0f : 0.0f;
    const float* p = props + ((size_t)b * NPROP + j) * 4;
    r0 = p[0] * nv; r1 = p[1] * nv; r2 = p[2] * nv; r3 = p[3] * nv;
  }

  const size_t roff = ((size_t)b * TT + t) * 4;
  rois[roff + 0] = r0; rois[roff + 1] = r1;
  rois[roff + 2] = r2; rois[roff + 3] = r3;
  cls[(size_t)b * TT + t] = c;
  deltas[roff + 0] = d0; deltas[roff + 1] = d1;
  deltas[roff + 2] = d2; deltas[roff + 3] = d3;
}

// =====================================================================
// Kernel 4: bilinear 28x28 crops (latency-bound 4-point gather from the
// 256 MB mask tensor, G innermost => 256 B stride).  800 blocks give
// ~200K in-flight gather threads to hide HBM latency; blocks with
// t >= P write the zero tail.
// =====================================================================
__global__ void k_masks(const float* __restrict__ props,
                        const float* __restrict__ gmasks,
                        const int*   __restrict__ posidx,
                        const float* __restrict__ posval,
                        const int*   __restrict__ assign_in,
                        float* __restrict__ masks) {
  const int blk = blockIdx.x;
  const int b = blk / TT;
  const int t = blk % TT;
  float* outp = masks + ((size_t)b * TT + t) * (MHH * MWW);

  if (t >= PP) {
    for (int p = threadIdx.x; p < MHH * MWW; p += blockDim.x) outp[p] = 0.0f;
    return;
  }

  __shared__ float sbox[4];
  __shared__ int   sg;
  __shared__ float sval;
  if (threadIdx.x == 0) {
    const int i = posidx[b * PP + t];
    const float* pb = props + ((size_t)b * NPROP + i) * 4;
    sbox[0] = pb[0]; sbox[1] = pb[1]; sbox[2] = pb[2]; sbox[3] = pb[3];
    sg   = assign_in[b * NPROP + i];
    sval = (posval[b * PP + t] >= 0.5f) ? 1.0f : 0.0f;
  }
  __syncthreads();

  const float y1 = sbox[0], x1 = sbox[1], y2 = sbox[2], x2 = sbox[3];
  const float valid = sval;
  const float* base = gmasks + ((size_t)b * HH * WW * NGT) + sg;

  for (int p = threadIdx.x; p < MHH * MWW; p += blockDim.x) {
    const int i = p / MWW;
    const int j = p % MWW;
    // match jnp: (y1 + (y2-y1)*i/(MH-1)) * (H-1)
    const float ys = (y1 + ((y2 - y1) * (float)i) / (float)(MHH - 1)) * (float)(HH - 1);
    const float xs = (x1 + ((x2 - x1) * (float)j) / (float)(MWW - 1)) * (float)(WW - 1);
    const float y0f = floorf(ys), x0f = floorf(xs);
    const float wy = ys - y0f,   wx = xs - x0f;
    const int y0i = (int)fminf(fmaxf(y0f,        0.0f), (float)(HH - 1));
    const int y1i = (int)fminf(fmaxf(y0f + 1.0f, 0.0f), (float)(HH - 1));
    const int x0i = (int)fminf(fmaxf(x0f,        0.0f), (float)(WW - 1));
    const int x1i = (int)fminf(fmaxf(x0f + 1.0f, 0.0f), (float)(WW - 1));

    const float* r0 = base + (size_t)y0i * (WW * NGT);
    const float* r1 = base + (size_t)y1i * (WW * NGT);
    __builtin_prefetch(r1 + (size_t)x0i * NGT, 0, 0);  // global_prefetch_b8
    const float v00 = r0[(size_t)x0i * NGT];
    const float v01 = r0[(size_t)x1i * NGT];
    const float v10 = r1[(size_t)x0i * NGT];
    const float v11 = r1[(size_t)x1i * NGT];

    const float top = v00 * (1.0f - wx) + v01 * wx;
    const float bot = v10 * (1.0f - wx) + v11 * wx;
    const float val = top * (1.0f - wy) + bot * wy;
    outp[p] = rintf(val) * valid;                      // jnp.round == RNE
  }
}

// =====================================================================
// Host-side launch
// =====================================================================
extern "C" void kernel_launch(void* const* d_in, const int* in_sizes, int n_in,
                              void* d_out, int out_size, void* d_ws, size_t ws_size,
                              hipStream_t stream) {
  (void)n_in; (void)out_size; (void)ws_size;
  const float* props  = (const float*)d_in[0];   // (B, 2000, 4) f32
  const int*   gids   = (const int*)  d_in[1];   // (B, 64) i32
  const float* gboxes = (const float*)d_in[2];   // (B, 64, 4) f32
  const float* gmasks = (const float*)d_in[3];   // (B, 512, 512, 64) f32
  const int B = in_sizes[1] / NGT;

  // workspace layout
  float* ws_iou    = (float*)d_ws;                              // B*N
  int*   ws_assign = (int*)(ws_iou + (size_t)B * NPROP);        // B*N
  int*   ws_posidx = ws_assign + (size_t)B * NPROP;             // B*P
  float* ws_posval = (float*)(ws_posidx + (size_t)B * PP);      // B*P
  int*   ws_negidx = (int*)(ws_posval + (size_t)B * PP);        // B*TN
  int*   ws_negn   = ws_negidx + (size_t)B * TN;                // B

  // output layout (concatenated flat, f32): rois | class_ids | deltas | masks
  float* out    = (float*)d_out;
  float* rois   = out;
  float* cls    = rois + (size_t)B * TT * 4;
  float* deltas = cls + (size_t)B * TT;
  float* masks  = deltas + (size_t)B * TT * 4;

  k_iou    <<<B, 256, 0, stream>>>(props, gids, gboxes, ws_iou, ws_assign);
  k_select <<<B, 256, 0, stream>>>(ws_iou, ws_posidx, ws_posval, ws_negidx, ws_negn);
  k_outputs<<<B, 256, 0, stream>>>(props, gids, gboxes, ws_posidx, ws_posval,
                                   ws_negidx, ws_negn, ws_assign, rois, cls, deltas);
  k_masks  <<<(unsigned)(B * TT), 256, 0, stream>>>(props, gmasks, ws_posidx,
                                                    ws_posval, ws_assign, masks);
}